// PeriodicKShellGraph_47519518163702
// MI455X (gfx1250) — compile-verified
//
#include <hip/hip_runtime.h>

// ---------------------------------------------------------------------------
// PeriodicKShellGraph on MI455X (gfx1250, wave32)
//
// d2(i,j) = |p_i|^2 + |q_j|^2 - 2 p_i.q_j
//         = [x y z 1] . [-2qx -2qy -2qz |q|^2]^T + |p|^2
// -> exact rank-4 f32 GEMM -> V_WMMA_F32_16X16X4_F32 per 16x16 tile.
//
// Roofline: inputs 24KB, outputs ~1.5MB (sub-us at 23.3 TB/s); the virtual
// 2048x55296 d2 matrix (450MB) never leaves LDS. Cost is dominated by top-32
// selection over ~55K candidates/atom -> 16K parallel selection lanes, each
// scanning ~6.9K candidates with a bitmask-filtered register insertion sort.
// Positions are staged into LDS with global_load_async_to_lds_b128 (ASYNCcnt).
// ---------------------------------------------------------------------------

typedef __attribute__((ext_vector_type(2))) float v2f;
typedef __attribute__((ext_vector_type(8))) float v8f;

#define KMAX    32
#define KSH     12
#define CUTOFF2 225.0f      // 15^2
#define TOLF    1.001f      // 1 + SHELL_TOL
#define BIGF    1.0e12f
#define NWAVES  8           // 256 threads/block
#define BATCH   3           // column-chunks (16 cols each) per barrier pair
#define NMAX    2048        // harness N (LDS staging capacity)

// --- setup: pos = frac @ lattice (packed float3), 27 cell shifts ------------
__global__ void kshell_setup(const float* __restrict__ frac,
                             const float* __restrict__ lat,
                             float4* __restrict__ shifts4,
                             float*  __restrict__ pos3,
                             int N) {
  int i = blockIdx.x * blockDim.x + threadIdx.x;
  float L00 = lat[0], L01 = lat[1], L02 = lat[2];
  float L10 = lat[3], L11 = lat[4], L12 = lat[5];
  float L20 = lat[6], L21 = lat[7], L22 = lat[8];
  if (i < N) {
    float f0 = frac[3*i+0], f1 = frac[3*i+1], f2 = frac[3*i+2];
    pos3[3*i+0] = f0*L00 + f1*L10 + f2*L20;
    pos3[3*i+1] = f0*L01 + f1*L11 + f2*L21;
    pos3[3*i+2] = f0*L02 + f1*L12 + f2*L22;
  }
  if (i < 27) {  // meshgrid(s,s,s,'ij'): cell c -> (c/9, (c/3)%3, c%3) - 1
    float cx = (float)(i / 9      ) - 1.0f;
    float cy = (float)((i / 3) % 3) - 1.0f;
    float cz = (float)(i % 3      ) - 1.0f;
    shifts4[i] = make_float4(cx*L00 + cy*L10 + cz*L20,
                             cx*L01 + cy*L11 + cz*L21,
                             cx*L02 + cy*L12 + cz*L22, 0.0f);
  }
}

// --- main: WMMA d2 tiles + bitmask top-32 + 8-way merge ---------------------
__launch_bounds__(32 * NWAVES)
__global__ void kshell_main(const float*  __restrict__ pos3,
                            const float4* __restrict__ shifts4,
                            int*   __restrict__ out_nbr,
                            float* __restrict__ out_dist,
                            float* __restrict__ out_vec,
                            float* __restrict__ out_mask,
                            int N) {
  // LDS: 24576 (pos) + 32768 (tiles U merge) + 432 + 64  ~= 58KB < 64KB cap
  __shared__ __align__(16) float s_pos[3 * NMAX];
  __shared__ __align__(16) union {
    float tile[NWAVES][BATCH][16][20];   // row stride 80B: 16B-aligned b128 rows
    struct { float d2[NWAVES][16][KMAX]; int ix[NWAVES][16][KMAX]; } m;
  } u;
  __shared__ float4 s_shift[27];
  __shared__ float  s_p2[16];

  const int tid  = threadIdx.x;
  const int wave = tid >> 5;
  const int lane = tid & 31;
  const int col  = lane & 15;
  const int hi   = lane >> 4;
  const int base = blockIdx.x * 16;

  // ---- stage positions into LDS via async-to-LDS (tracked by ASYNCcnt) ----
  const int nchunk16 = (3 * N) >> 2;                // float4 chunks of pos3
#if defined(__AMDGCN__)
  {
    // LDS aperture maps flat addr[31:0] directly to the LDS byte offset.
    unsigned lbase = (unsigned)(unsigned long long)(const void*)&s_pos[0];
    unsigned long long gbase = (unsigned long long)(const void*)pos3;
    for (int i = tid; i < nchunk16; i += 32 * NWAVES) {
      unsigned           laddr = lbase + 16u * (unsigned)i;
      unsigned long long gaddr = gbase + 16ull * (unsigned)i;
      asm volatile("global_load_async_to_lds_b128 %0, %1, off"
                   :: "v"(laddr), "v"(gaddr)
                   : "memory");
    }
    asm volatile("s_wait_asynccnt 0" ::: "memory");
  }
#else
  for (int i = tid; i < nchunk16; i += 32 * NWAVES)
    ((float4*)s_pos)[i] = ((const float4*)pos3)[i];
#endif
  if (tid < 27) s_shift[tid] = shifts4[tid];
  const float px = pos3[3*(base+col)+0];
  const float py = pos3[3*(base+col)+1];
  const float pz = pos3[3*(base+col)+2];
  if (tid < 16) s_p2[tid] = px*px + py*py + pz*pz;
  __syncthreads();

  // A (16x4): row m = [x,y,z,1]; lanes 0-15 hold (K0,K1), 16-31 hold (K2,K3)
  v2f A;
  A.x = hi ? pz   : px;
  A.y = hi ? 1.0f : py;
  // C (16x16): C[m][n] = |p_m|^2 ; VGPR r holds row m = r + 8*hi
  v8f Cin;
#pragma unroll
  for (int r = 0; r < 8; ++r) Cin[r] = s_p2[r + hi*8];

  // running top-KMAX (ascending d2), one atom per owner lane (0..15)
  float bd[KMAX];
  int   bi[KMAX];
#pragma unroll
  for (int k = 0; k < KMAX; ++k) { bd[k] = BIGF; bi[k] = k; }
  const int selfFlat = 13 * N + (base + col);
  const float CUT_UP = __uint_as_float(0x43610001u);  // nextafter(225.f,+inf)

  const int chunksTotal = (27 * N) >> 4;
  const int perWave     = (chunksTotal + NWAVES - 1) / NWAVES;
  const int iters       = (perWave + BATCH - 1) / BATCH;
  const int c0          = wave * perWave;

  // incremental (flat0, jbase, cell): one divide per wave, +16/wrap per chunk
  int flat0 = c0 << 4;
  int cell  = flat0 / N;                    // N%16==0 -> cell constant in chunk
  int jbase = flat0 - cell * N;

  for (int it = 0; it < iters; ++it) {
    int  f0[BATCH];
    bool act[BATCH];                        // wave-uniform activity flags
#pragma unroll
    for (int b = 0; b < BATCH; ++b) {
      const int cw = it * BATCH + b;
      act[b] = (cw < perWave) && (c0 + cw < chunksTotal);
      f0[b]  = flat0;
      if (act[b]) {
        const int j = jbase + col;
        float4 sh = s_shift[cell];
        float qx = s_pos[3*j+0] + sh.x;
        float qy = s_pos[3*j+1] + sh.y;
        float qz = s_pos[3*j+2] + sh.z;
        float q2 = qx*qx + qy*qy + qz*qz;
        v2f B;                               // col n = [-2qx,-2qy,-2qz,|q|^2]
        B.x = hi ? -2.0f*qz : -2.0f*qx;
        B.y = hi ?  q2      : -2.0f*qy;
        v8f D = __builtin_amdgcn_wmma_f32_16x16x4_f32(
            false, A, false, B, (short)0, Cin, false, false);
#pragma unroll
        for (int r = 0; r < 8; ++r)
          u.tile[wave][b][r + hi*8][col] = D[r];
      }
      flat0 += 16;                           // advance after use (tail harmless)
      jbase += 16;
      if (jbase >= N) { jbase = 0; cell += 1; }
    }
    __syncthreads();

    if (lane < 16) {
#pragma unroll
      for (int b = 0; b < BATCH; ++b) {
        if (!act[b]) continue;
        // whole row in 4x ds_load_b128, single wait
        const float4* rp = (const float4*)&u.tile[wave][b][lane][0];
        float4 r0 = rp[0], r1 = rp[1], r2 = rp[2], r3 = rp[3];
        const float cand[16] = { r0.x, r0.y, r0.z, r0.w,
                                 r1.x, r1.y, r1.z, r1.w,
                                 r2.x, r2.y, r2.z, r2.w,
                                 r3.x, r3.y, r3.z, r3.w };
        // one compare per candidate: thr subsumes cutoff AND current worst
        const float thr = fminf(bd[KMAX-1], CUT_UP);
        unsigned pass = 0u;
#pragma unroll
        for (int t = 0; t < 16; ++t)
          pass |= (cand[t] < thr) ? (1u << t) : 0u;
        const int st = selfFlat - f0[b];
        if ((unsigned)st < 16u) pass &= ~(1u << st);   // drop self image

        while (pass) {                       // rare path (~2 bits per 16)
          const int t = __builtin_ctz(pass);
          pass &= pass - 1;
          const float d2 = u.tile[wave][b][lane][t];
          if (d2 < bd[KMAX-1]) {             // recheck vs updated worst
            bd[KMAX-1] = d2; bi[KMAX-1] = f0[b] + t;
#pragma unroll
            for (int k = KMAX-1; k > 0; --k) {   // one bubble pass
              if (bd[k] < bd[k-1]) {
                float td = bd[k]; bd[k] = bd[k-1]; bd[k-1] = td;
                int   ti = bi[k]; bi[k] = bi[k-1]; bi[k-1] = ti;
              }
            }
          }
        }
      }
    }
    __syncthreads();
  }

  // ---- dump per-wave sorted lists (tile buffers dead -> reuse as merge) ----
  if (lane < 16) {
#pragma unroll
    for (int k = 0; k < KMAX; ++k) {
      u.m.d2[wave][lane][k] = bd[k];
      u.m.ix[wave][lane][k] = bi[k];
    }
  }
  __syncthreads();

  // ---- wave 0, lanes 0..15: 8-way merge -> global top-32, emit outputs ----
  if (wave == 0 && lane < 16) {
    int head[NWAVES];
#pragma unroll
    for (int w = 0; w < NWAVES; ++w) head[w] = 0;

    float md[KMAX];
    int   mi[KMAX];
#pragma unroll
    for (int k = 0; k < KMAX; ++k) {
      // at read time sum(head)==k<=31, so every head[w]<=31: in-bounds
      float best = u.m.d2[0][lane][head[0]];
      int   bw = 0, bh = head[0];
#pragma unroll
      for (int w = 1; w < NWAVES; ++w) {
        int   h = head[w];
        float v = u.m.d2[w][lane][h];
        if (v < best) { best = v; bw = w; bh = h; }
      }
      md[k] = best;
      mi[k] = u.m.ix[bw][lane][bh];
#pragma unroll
      for (int w = 0; w < NWAVES; ++w)       // no dynamic reg indexing
        if (w == bw) head[w] = head[w] + 1;
    }

    const int atom = base + lane;

    // r_shell: dist of 12th neighbor, recomputed from positions (as reference)
    float rshell;
    {
      int flat = mi[KSH-1];
      int cc   = flat / N;
      int j    = flat - cc * N;
      float4 sh = s_shift[cc];
      float vx = s_pos[3*j+0] + sh.x - px;
      float vy = s_pos[3*j+1] + sh.y - py;
      float vz = s_pos[3*j+2] + sh.z - pz;
      rshell = sqrtf(fmaxf(vx*vx + vy*vy + vz*vz, 1e-12f));
    }

#pragma unroll 1
    for (int k = 0; k < KMAX; ++k) {
      int flat = mi[k];
      int cc   = flat / N;
      int j    = flat - cc * N;
      float4 sh = s_shift[cc];
      float vx = s_pos[3*j+0] + sh.x - px;
      float vy = s_pos[3*j+1] + sh.y - py;
      float vz = s_pos[3*j+2] + sh.z - pz;
      float dist = sqrtf(fmaxf(vx*vx + vy*vy + vz*vz, 1e-12f));
      bool  m    = (md[k] < BIGF) && (dist <= rshell * TOLF);
      int   o    = atom * KMAX + k;
      out_nbr [o]       = j;
      out_dist[o]       = m ? dist : 0.0f;
      out_vec [o*3 + 0] = m ? vx : 0.0f;
      out_vec [o*3 + 1] = m ? vy : 0.0f;
      out_vec [o*3 + 2] = m ? vz : 0.0f;
      out_mask[o]       = m ? 1.0f : 0.0f;   // bool emitted as 0/1 element
    }
  }
}

// ---------------------------------------------------------------------------
extern "C" void kernel_launch(void* const* d_in, const int* in_sizes, int n_in,
                              void* d_out, int out_size, void* d_ws, size_t ws_size,
                              hipStream_t stream) {
  const float* frac = (const float*)d_in[0];   // (N,3) f32
  const float* lat  = (const float*)d_in[1];   // (3,3) f32
  const int N = in_sizes[0] / 3;               // 2048 (multiple of 16, <= NMAX)

  float4* shifts4 = (float4*)d_ws;             // 27 float4 (432B, 16B-aligned)
  float*  pos3    = (float*)d_ws + 27 * 4;     // N packed float3

  // outputs concatenated flat in return order, 4-byte elements each:
  // nbr_j (N*32 i32) | dist (N*32 f32) | vec (N*32*3 f32) | mask (N*32)
  float* out      = (float*)d_out;
  int*   out_nbr  = (int*)out;
  float* out_dist = out + (size_t)N * KMAX;
  float* out_vec  = out + (size_t)2 * N * KMAX;
  float* out_mask = out + (size_t)5 * N * KMAX;

  kshell_setup<<<(N + 255) / 256, 256, 0, stream>>>(frac, lat, shifts4, pos3, N);
  kshell_main<<<N / 16, 32 * NWAVES, 0, stream>>>(pos3, shifts4,
                                                  out_nbr, out_dist, out_vec,
                                                  out_mask, N);
}